// SS2D_74345883894589
// MI455X (gfx1250) — compile-verified
//
#include <hip/hip_runtime.h>
#include <math.h>

// ---------------------------------------------------------------------------
// SS2D for MI455X (gfx1250, wave32). fp32 WMMA (v_wmma_f32_16x16x4_f32) for
// all GEMMs; chunked parallel selective scan (depth 64 instead of 4096).
// ---------------------------------------------------------------------------

typedef __attribute__((ext_vector_type(2))) float v2f;
typedef __attribute__((ext_vector_type(8))) float v8f;

#define DMODEL 96
#define DSTATE 16
#define DINNER 192
#define DTRANK 6
#define NB     4
#define HH     64
#define WWID   64
#define LLEN   4096       // HH*WWID
#define KDIR   4
#define CPROJ  38         // DTRANK + 2*DSTATE
#define NCHUNK 64
#define CHLEN  64         // LLEN / NCHUNK
#define XZW    384        // 2*DINNER

// workspace byte offsets
#define OFF_XZ   ((size_t)0)                       // 16384*384 f32
#define OFF_XC   (OFF_XZ  + (size_t)16384*384*4)   // (B,D,L) f32
#define OFF_DBL  (OFF_XC  + (size_t)NB*DINNER*LLEN*4)        // (K,B,L,38)
#define OFF_EB   (OFF_DBL + (size_t)KDIR*NB*LLEN*CPROJ*4)    // (B,K,D,ch,16)
#define OFF_HB   (OFF_EB  + (size_t)NB*KDIR*DINNER*NCHUNK*16*4)
#define OFF_Y    (OFF_HB  + (size_t)NB*KDIR*DINNER*NCHUNK*16*4)  // (B,D,L)
#define OFF_G    (OFF_Y   + (size_t)NB*DINNER*LLEN*4)            // (B,L,D)

__device__ __forceinline__ float sigmoidf_(float x) { return 1.f / (1.f + __expf(-x)); }
__device__ __forceinline__ float softplusf_(float x) { return x > 20.f ? x : log1pf(__expf(x)); }

// scan-order position t -> spatial index l for direction k (involution: same
// map serves the gather of xs and the scatter of out_y back to (h,w)).
__device__ __forceinline__ int scan_idx(int k, int t) {
  switch (k) {
    case 0:  return t;
    case 1:  return ((t & 63) << 6) | (t >> 6);
    case 2:  return (LLEN - 1) - t;
    default: { int j = (LLEN - 1) - t; return ((j & 63) << 6) | (j >> 6); }
  }
}

__device__ __forceinline__ v8f wmma4(v2f a, v2f b, v8f c) {
  // D = A(16x4) * B(4x16) + C ; fp32 WMMA, wave32
  return __builtin_amdgcn_wmma_f32_16x16x4_f32(false, a, false, b, (short)0, c, false, false);
}

// ---------------- 1) in_proj: xz[16384][384] = x[16384][96] @ W^T ----------
__global__ void k_inproj(const float* __restrict__ x, const float* __restrict__ w,
                         float* __restrict__ xz) {
  const int lane = threadIdx.x & 31;
  const int wave = threadIdx.x >> 5;
  const int m0 = (blockIdx.x * 4 + wave) * 16;
  const int n0 = blockIdx.y * 16;
  const int r  = lane & 15;          // A row / B col
  const int kb = (lane >> 4) * 2;    // K sub-pair per ISA 32-bit A layout
  v8f acc = {};
  for (int k0 = 0; k0 < DMODEL; k0 += 4) {
    v2f a, b;
    a.x = x[(size_t)(m0 + r) * DMODEL + k0 + kb];
    a.y = x[(size_t)(m0 + r) * DMODEL + k0 + kb + 1];
    b.x = w[(size_t)(n0 + r) * DMODEL + k0 + kb];
    b.y = w[(size_t)(n0 + r) * DMODEL + k0 + kb + 1];
    acc = wmma4(a, b, acc);
  }
#pragma unroll
  for (int i = 0; i < 8; ++i) {
    int ro = m0 + i + (lane >> 4) * 8;
    xz[(size_t)ro * XZW + n0 + (lane & 15)] = acc[i];
  }
}

// ---------------- 2) depthwise 3x3 conv + bias + SiLU ----------------------
__global__ void k_conv(const float* __restrict__ xz, const float* __restrict__ cw,
                       const float* __restrict__ cb, float* __restrict__ xc) {
  int gid = blockIdx.x * blockDim.x + threadIdx.x;  // ((b*192+d)*4096)+h*64+w
  int w = gid & 63, h = (gid >> 6) & 63;
  int d = (gid >> 12) % DINNER, b = (gid >> 12) / DINNER;
  float s = 0.f;
#pragma unroll
  for (int dh = -1; dh <= 1; ++dh)
#pragma unroll
    for (int dw = -1; dw <= 1; ++dw) {
      int hh = h + dh, ww = w + dw;
      if (hh >= 0 && hh < HH && ww >= 0 && ww < WWID)
        s += xz[((size_t)(b << 12) + (hh << 6) + ww) * XZW + d] *
             cw[d * 9 + (dh + 1) * 3 + (dw + 1)];
    }
  s += cb[d];
  s = s * sigmoidf_(s);  // SiLU
  xc[((size_t)b * DINNER + d) * LLEN + (h << 6) + w] = s;
}

// ---------------- 3) x_dbl[k] = xs[k]^T @ x_proj_w[k]^T (gather on the fly)
__global__ void k_xdbl(const float* __restrict__ xc, const float* __restrict__ xpw,
                       float* __restrict__ dbl) {
  const int lane = threadIdx.x & 31;
  const int wave = threadIdx.x >> 5;
  const int k  = blockIdx.z;
  const int m0 = (blockIdx.x * 4 + wave) * 16;
  const int n0 = blockIdx.y * 16;
  const int r  = lane & 15;
  const int hk = lane >> 4;

  const int rowa = m0 + r;
  const int b = rowa >> 12, t = rowa & (LLEN - 1);
  const int l = scan_idx(k, t);
  const float* xcb = xc + (size_t)b * DINNER * LLEN + l;
  const int c = n0 + r;
  const bool cok = (c < CPROJ);

  v8f acc = {};
  for (int k0 = 0; k0 < DINNER; k0 += 4) {
    int d0 = k0 + hk * 2;
    v2f a, bv;
    a.x = xcb[(size_t)d0 * LLEN];
    a.y = xcb[(size_t)(d0 + 1) * LLEN];
    bv.x = cok ? xpw[((size_t)k * CPROJ + c) * DINNER + d0] : 0.f;
    bv.y = cok ? xpw[((size_t)k * CPROJ + c) * DINNER + d0 + 1] : 0.f;
    acc = wmma4(a, bv, acc);
  }
#pragma unroll
  for (int i = 0; i < 8; ++i) {
    int ro = m0 + i + hk * 8;
    int co = n0 + (lane & 15);
    if (co < CPROJ) {
      int bb = ro >> 12, tt = ro & (LLEN - 1);
      dbl[((size_t)(k * NB + bb) * LLEN + tt) * CPROJ + co] = acc[i];
    }
  }
}

// ---------------- 4a) scan pass 1: per-chunk (prod E, partial h) -----------
__global__ void k_scan1(const float* __restrict__ xc, const float* __restrict__ dbl,
                        const float* __restrict__ dtw, const float* __restrict__ dtb,
                        const float* __restrict__ alog,
                        float* __restrict__ Eb, float* __restrict__ Hb) {
  int gid = blockIdx.x * blockDim.x + threadIdx.x;  // ((b*4+k)*192+d)*64 + c
  int c = gid & 63;
  int d = (gid >> 6) % DINNER;
  int bk = gid / (NCHUNK * DINNER);
  int k = bk & 3, b = bk >> 2;

  float A[DSTATE], hS[DSTATE], ES[DSTATE], w6[DTRANK];
#pragma unroll
  for (int n = 0; n < DSTATE; ++n) {
    A[n] = -__expf(alog[(((size_t)k * DINNER + d) << 4) + n]);
    hS[n] = 0.f; ES[n] = 1.f;
  }
#pragma unroll
  for (int r = 0; r < DTRANK; ++r) w6[r] = dtw[((size_t)k * DINNER + d) * DTRANK + r];
  const float bias = dtb[k * DINNER + d];
  const float* xcd = xc + ((size_t)b * DINNER + d) * LLEN;
  const float* dp  = dbl + ((size_t)(k * NB + b) * LLEN + c * CHLEN) * CPROJ;

  for (int tt = 0; tt < CHLEN; ++tt) {
    const int t = c * CHLEN + tt;
    const float* row = dp + (size_t)tt * CPROJ;
    float raw = bias;
#pragma unroll
    for (int r = 0; r < DTRANK; ++r) raw += row[r] * w6[r];
    const float dt = softplusf_(raw);
    const float u  = xcd[scan_idx(k, t)];
    const float du = dt * u;
#pragma unroll
    for (int n = 0; n < DSTATE; ++n) {
      float e = __expf(dt * A[n]);
      hS[n] = e * hS[n] + du * row[DTRANK + n];
      ES[n] *= e;
    }
  }
  size_t o = (size_t)gid * DSTATE;
#pragma unroll
  for (int n = 0; n < DSTATE; ++n) { Eb[o + n] = ES[n]; Hb[o + n] = hS[n]; }
}

// ---------------- 4b) prefix over chunks (in-place: Hb becomes h_init) ----
__global__ void k_prefix(const float* __restrict__ Eb, float* __restrict__ Hb) {
  int gid = blockIdx.x * blockDim.x + threadIdx.x;  // (b*4+k)*192+d)*16 + n
  int n = gid & 15;
  size_t bkd = (size_t)(gid >> 4);
  float run = 0.f;
  for (int c = 0; c < NCHUNK; ++c) {
    size_t idx = (bkd * NCHUNK + c) * DSTATE + n;
    float hp = Hb[idx];
    float e  = Eb[idx];
    Hb[idx] = run;            // h_init for this chunk
    run = e * run + hp;
  }
}

__global__ void k_zero(float* __restrict__ p) {
  p[blockIdx.x * blockDim.x + threadIdx.x] = 0.f;
}

// ---------------- 4c) scan pass 2: replay with h_init, scatter y ----------
__global__ void k_scan2(const float* __restrict__ xc, const float* __restrict__ dbl,
                        const float* __restrict__ dtw, const float* __restrict__ dtb,
                        const float* __restrict__ alog, const float* __restrict__ Ds,
                        const float* __restrict__ Hb, float* __restrict__ y) {
  int gid = blockIdx.x * blockDim.x + threadIdx.x;
  int c = gid & 63;
  int d = (gid >> 6) % DINNER;
  int bk = gid / (NCHUNK * DINNER);
  int k = bk & 3, b = bk >> 2;

  float A[DSTATE], hS[DSTATE], w6[DTRANK];
  const size_t o = (size_t)gid * DSTATE;
#pragma unroll
  for (int n = 0; n < DSTATE; ++n) {
    A[n] = -__expf(alog[(((size_t)k * DINNER + d) << 4) + n]);
    hS[n] = Hb[o + n];
  }
#pragma unroll
  for (int r = 0; r < DTRANK; ++r) w6[r] = dtw[((size_t)k * DINNER + d) * DTRANK + r];
  const float bias = dtb[k * DINNER + d];
  const float Dval = Ds[k * DINNER + d];
  const float* xcd = xc + ((size_t)b * DINNER + d) * LLEN;
  const float* dp  = dbl + ((size_t)(k * NB + b) * LLEN + c * CHLEN) * CPROJ;
  float* yd = y + ((size_t)b * DINNER + d) * LLEN;

  for (int tt = 0; tt < CHLEN; ++tt) {
    const int t = c * CHLEN + tt;
    const float* row = dp + (size_t)tt * CPROJ;
    float raw = bias;
#pragma unroll
    for (int r = 0; r < DTRANK; ++r) raw += row[r] * w6[r];
    const float dt = softplusf_(raw);
    const int l = scan_idx(k, t);
    const float u = xcd[l];
    const float du = dt * u;
    float yv = 0.f;
#pragma unroll
    for (int n = 0; n < DSTATE; ++n) {
      float e = __expf(dt * A[n]);
      hS[n] = e * hS[n] + du * row[DTRANK + n];
      yv += hS[n] * row[DTRANK + DSTATE + n];
    }
    atomicAdd(&yd[l], yv + Dval * u);   // merge 4 directions
  }
}

// ---------------- 5) LayerNorm(D=192) + SiLU(z) gate -> g (B,L,D) ---------
__global__ void k_lngate(const float* __restrict__ y, const float* __restrict__ xz,
                         const float* __restrict__ lnw, const float* __restrict__ lnb,
                         float* __restrict__ g) {
  __shared__ float ssum[64], ssum2[64];
  const int tid = threadIdx.x;            // 64 threads, 3 channels each
  const int b = blockIdx.x >> 12;
  const int l = blockIdx.x & (LLEN - 1);
  float v[3], s = 0.f, s2 = 0.f;
#pragma unroll
  for (int i = 0; i < 3; ++i) {
    int d = tid + i * 64;
    v[i] = y[((size_t)b * DINNER + d) * LLEN + l];
    s += v[i]; s2 += v[i] * v[i];
  }
  ssum[tid] = s; ssum2[tid] = s2;
  __syncthreads();
  for (int off = 32; off >= 1; off >>= 1) {
    if (tid < off) { ssum[tid] += ssum[tid + off]; ssum2[tid] += ssum2[tid + off]; }
    __syncthreads();
  }
  const float mu = ssum[0] * (1.f / DINNER);
  const float var = ssum2[0] * (1.f / DINNER) - mu * mu;
  const float rs = rsqrtf(var + 1e-5f);
#pragma unroll
  for (int i = 0; i < 3; ++i) {
    int d = tid + i * 64;
    float norm = (v[i] - mu) * rs * lnw[d] + lnb[d];
    float z = xz[((size_t)(b << 12) + l) * XZW + DINNER + d];
    g[((size_t)(b << 12) + l) * DINNER + d] = norm * (z * sigmoidf_(z));
  }
}

// ---------------- 6) out_proj: out[16384][96] = g @ out_proj_w^T ----------
__global__ void k_outproj(const float* __restrict__ g, const float* __restrict__ w,
                          float* __restrict__ out) {
  const int lane = threadIdx.x & 31;
  const int wave = threadIdx.x >> 5;
  const int m0 = (blockIdx.x * 4 + wave) * 16;
  const int n0 = blockIdx.y * 16;
  const int r  = lane & 15;
  const int kb = (lane >> 4) * 2;
  v8f acc = {};
  for (int k0 = 0; k0 < DINNER; k0 += 4) {
    v2f a, b;
    a.x = g[(size_t)(m0 + r) * DINNER + k0 + kb];
    a.y = g[(size_t)(m0 + r) * DINNER + k0 + kb + 1];
    b.x = w[(size_t)(n0 + r) * DINNER + k0 + kb];
    b.y = w[(size_t)(n0 + r) * DINNER + k0 + kb + 1];
    acc = wmma4(a, b, acc);
  }
#pragma unroll
  for (int i = 0; i < 8; ++i) {
    int ro = m0 + i + (lane >> 4) * 8;
    out[(size_t)ro * DMODEL + n0 + (lane & 15)] = acc[i];
  }
}

// ---------------------------------------------------------------------------
extern "C" void kernel_launch(void* const* d_in, const int* in_sizes, int n_in,
                              void* d_out, int out_size, void* d_ws, size_t ws_size,
                              hipStream_t stream) {
  (void)in_sizes; (void)n_in; (void)out_size; (void)ws_size;
  const float* x    = (const float*)d_in[0];
  const float* ipw  = (const float*)d_in[1];
  const float* cw   = (const float*)d_in[2];
  const float* cb   = (const float*)d_in[3];
  const float* xpw  = (const float*)d_in[4];
  const float* dtw  = (const float*)d_in[5];
  const float* dtb  = (const float*)d_in[6];
  const float* alog = (const float*)d_in[7];
  const float* Ds   = (const float*)d_in[8];
  const float* lnw  = (const float*)d_in[9];
  const float* lnb  = (const float*)d_in[10];
  const float* opw  = (const float*)d_in[11];
  float* out = (float*)d_out;

  char* ws = (char*)d_ws;
  float* xz  = (float*)(ws + OFF_XZ);
  float* xc  = (float*)(ws + OFF_XC);
  float* dbl = (float*)(ws + OFF_DBL);
  float* Eb  = (float*)(ws + OFF_EB);
  float* Hb  = (float*)(ws + OFF_HB);
  float* y   = (float*)(ws + OFF_Y);
  float* g   = (float*)(ws + OFF_G);

  // 1) in_proj GEMM (WMMA): 16384 x 384 x 96
  k_inproj<<<dim3(256, 24), dim3(128), 0, stream>>>(x, ipw, xz);
  // 2) depthwise conv + SiLU
  k_conv<<<dim3(12288), dim3(256), 0, stream>>>(xz, cw, cb, xc);
  // 3) x_dbl GEMMs (WMMA, per direction): 16384 x 48(pad of 38) x 192
  k_xdbl<<<dim3(256, 3, KDIR), dim3(128), 0, stream>>>(xc, xpw, dbl);
  // 4) chunked selective scan
  k_scan1<<<dim3(768), dim3(256), 0, stream>>>(xc, dbl, dtw, dtb, alog, Eb, Hb);
  k_prefix<<<dim3(192), dim3(256), 0, stream>>>(Eb, Hb);
  k_zero<<<dim3(12288), dim3(256), 0, stream>>>(y);  // 4*192*4096 floats
  k_scan2<<<dim3(768), dim3(256), 0, stream>>>(xc, dbl, dtw, dtb, alog, Ds, Hb, y);
  // 5) LayerNorm + SiLU gate
  k_lngate<<<dim3(16384), dim3(64), 0, stream>>>(y, xz, lnw, lnb, g);
  // 6) out_proj GEMM (WMMA): 16384 x 96 x 192
  k_outproj<<<dim3(256, 6), dim3(128), 0, stream>>>(g, opw, out);
}